// Attention_42202348650848
// MI455X (gfx1250) — compile-verified
//
#include <hip/hip_runtime.h>
#include <hip/hip_bf16.h>
#include <math.h>

// Problem constants (from reference setup_inputs)
constexpr int B = 32;
constexpr int C = 1024;
constexpr int Q = 128;
constexpr int D = 768;

typedef float v2f __attribute__((ext_vector_type(2)));
typedef float v8f __attribute__((ext_vector_type(8)));
typedef unsigned int u32;
typedef u32 u32x4 __attribute__((ext_vector_type(4)));
typedef int  i32x4 __attribute__((ext_vector_type(4)));
typedef int  i32x8 __attribute__((ext_vector_type(8)));

// CDNA5 fp32 WMMA: D(16x16) = A(16x4) * B(4x16) + C
__device__ __forceinline__ v8f wmma_f32(v2f a, v2f b, v8f c) {
    return __builtin_amdgcn_wmma_f32_16x16x4_f32(
        false, a, false, b, (short)0, c, false, false);
}

// ---- Tensor Data Mover: 2D tile load (global -> LDS), D# per ISA ch.8 ------
// tile = tile_d1 rows x tile_d0 contiguous elements (fp32), row stride
// stride_d0 elements. lds_off = byte offset of destination in workgroup LDS.
__device__ __forceinline__ void tdm_load_2d(const void* gaddr, u32 lds_off,
                                            u32 tensor_d0, u32 tensor_d1,
                                            u32 tile_d0, u32 tile_d1,
                                            u32 stride_d0) {
    unsigned long long ga = (unsigned long long)(size_t)gaddr;
    u32x4 g0;
    g0[0] = 1u;                                   // count=1 (valid user D#)
    g0[1] = lds_off;                              // lds_addr (bytes)
    g0[2] = (u32)(ga & 0xffffffffull);            // global_addr[31:0]
    g0[3] = (u32)((ga >> 32) & 0x01ffffffull)     // global_addr[56:32]
          | (2u << 30);                           // type=2 ("image")
    i32x8 g1;
    g1[0] = (int)(2u << 16);                      // data_size=2 -> 4 bytes
    g1[1] = (int)((tensor_d0 & 0xffffu) << 16);   // tensor_dim0[15:0]
    g1[2] = (int)((tensor_d0 >> 16) | ((tensor_d1 & 0xffffu) << 16));
    g1[3] = (int)((tensor_d1 >> 16) | ((tile_d0 & 0xffffu) << 16));
    g1[4] = (int)(tile_d1 & 0xffffu);             // tile_dim1 (tile_dim2=0)
    g1[5] = (int)stride_d0;                       // tensor_dim0_stride[31:0]
    g1[6] = 0;
    g1[7] = 0;
    i32x4 z4 = {0, 0, 0, 0};
#if __clang_major__ >= 23
    i32x8 z8 = {0, 0, 0, 0, 0, 0, 0, 0};
    __builtin_amdgcn_tensor_load_to_lds(g0, g1, z4, z4, z8, 0);
#else
    __builtin_amdgcn_tensor_load_to_lds(g0, g1, z4, z4, 0);
#endif
}

#define NEG_INF (-__builtin_inff())
#define POS_INF (__builtin_inff())

// ---------------- workspace layout (floats) ----------------
constexpr size_t OFF_QW  = 0;                         // B*Q*D
constexpr size_t OFF_XQ  = OFF_QW  + (size_t)B*Q*D;   // B*Q
constexpr size_t OFF_XC  = OFF_XQ  + (size_t)B*Q;     // B*C
constexpr size_t OFF_RM  = OFF_XC  + (size_t)B*C;     // B*C
constexpr size_t OFF_W   = OFF_RM  + (size_t)B*C;     // B*C
constexpr size_t OFF_Q2C = OFF_W   + (size_t)B*C;     // B*D
constexpr size_t OFF_SIM = OFF_Q2C + (size_t)B*D;     // B*C*Q

// ---------------- stage 1a: qw = query * wqc ; x_q = query . wq -------------
__global__ void __launch_bounds__(256) k_qw_xq(
    const float* __restrict__ query, const float* __restrict__ wq,
    const float* __restrict__ wqc, float* __restrict__ qw,
    float* __restrict__ xq) {
    const int row = blockIdx.x;              // b*Q + q
    const int tid = threadIdx.x;
    const float* qrow = query + (size_t)row * D;
    float* qwrow = qw + (size_t)row * D;
    float partial = 0.f;
    for (int d = tid; d < D; d += 256) {
        float v = qrow[d];
        qwrow[d] = v * wqc[d];
        partial += v * wq[d];
    }
    __shared__ float red[256];
    red[tid] = partial;
    __syncthreads();
    for (int s = 128; s > 0; s >>= 1) {
        if (tid < s) red[tid] += red[tid + s];
        __syncthreads();
    }
    if (tid == 0) xq[row] = red[0];
}

// ---------------- stage 1b: x_c = context . wc ------------------------------
__global__ void __launch_bounds__(256) k_xc(
    const float* __restrict__ context, const float* __restrict__ wc,
    float* __restrict__ xc) {
    const int row = blockIdx.x;              // b*C + c
    const int tid = threadIdx.x;
    const float* crow = context + (size_t)row * D;
    float partial = 0.f;
    for (int d = tid; d < D; d += 256) partial += crow[d] * wc[d];
    __shared__ float red[256];
    red[tid] = partial;
    __syncthreads();
    for (int s = 128; s > 0; s >>= 1) {
        if (tid < s) red[tid] += red[tid + s];
        __syncthreads();
    }
    if (tid == 0) xc[row] = red[0];
}

// ---------------- stage 2: sim[b,c,q] via TDM-staged WMMA -------------------
// grid: (C/128, B). 8 waves, each wave owns 16 C-rows x 128 Q cols (8 tiles).
// Per 64-wide K chunk: TDM stages context tile (128x64) and qw tile (128x64)
// into LDS once per workgroup; inner loop is ds_load_b64 + v_wmma only.
__global__ void __launch_bounds__(256) k_sim(
    const float* __restrict__ context, const float* __restrict__ qw,
    const float* __restrict__ xq, const float* __restrict__ xc,
    const int* __restrict__ cmask, const int* __restrict__ qmask,
    float* __restrict__ sim, float* __restrict__ rowmax) {
    __shared__ float ldsT[2 * 128 * 64];     // [0]=A (context), [1]=B (qw)
    float* ldsA = ldsT;
    float* ldsB = ldsT + 128 * 64;

    const int b    = blockIdx.y;
    const int wave = threadIdx.x >> 5;
    const int lane = threadIdx.x & 31;
    const int half = lane >> 4;              // K half-select
    const int lr   = lane & 15;              // M (A) / N (B) within tile
    const int cbase = blockIdx.x * 128 + wave * 16;

    const float* Actx = context + ((size_t)(b * C) + blockIdx.x * 128) * D;
    const float* Bqw  = qw + (size_t)(b * Q) * D;
    const u32 ldsA_off = (u32)(size_t)(void*)ldsA;
    const u32 ldsB_off = (u32)(size_t)(void*)ldsB;

    v8f acc[8] = {};
    for (int kc = 0; kc < D / 64; ++kc) {
        if (wave == 0) {
            tdm_load_2d(Actx + kc * 64, ldsA_off, (u32)D, 128u, 64u, 128u, (u32)D);
            tdm_load_2d(Bqw  + kc * 64, ldsB_off, (u32)D, 128u, 64u, 128u, (u32)D);
            __builtin_amdgcn_s_wait_tensorcnt(0);
        }
        __syncthreads();
        // A frag: row = wave*16+lr, K = kk + 2*half + {0,1}
        const float* Ap = ldsA + (wave * 16 + lr) * 64 + 2 * half;
        // B frag: col q = t*16+lr, same K split
        const float* Bp = ldsB + lr * 64 + 2 * half;
        for (int kk = 0; kk < 64; kk += 4) {
            v2f a = *(const v2f*)(Ap + kk);
#pragma unroll
            for (int t = 0; t < 8; ++t) {
                v2f bf = *(const v2f*)(Bp + t * 16 * 64 + kk);
                acc[t] = wmma_f32(a, bf, acc[t]);
            }
        }
        __syncthreads();
    }

    // C/D layout: element (reg r, lane) -> M = r + 8*half, N = lr
    float rm[8];
#pragma unroll
    for (int r = 0; r < 8; ++r) rm[r] = NEG_INF;

#pragma unroll
    for (int t = 0; t < 8; ++t) {
        const int q   = t * 16 + lr;
        const float xqv = xq[b * Q + q];
        const float mq  = (qmask[b * Q + q] == 1) ? 1.0f : NEG_INF;
#pragma unroll
        for (int r = 0; r < 8; ++r) {
            const int c = cbase + 8 * half + r;
            float s = acc[t][r] + xqv + xc[b * C + c];
            const float mc = (cmask[b * C + c] == 1) ? 1.0f : NEG_INF;
            s = s * mc;          // faithful inf-multiply masking
            s = s * mq;
            if (s == POS_INF) s = NEG_INF;
            sim[((size_t)(b * C) + c) * Q + q] = s;
            rm[r] = fmaxf(rm[r], s);
        }
    }

    // reduce row max across the 16 lanes holding the same rows
#pragma unroll
    for (int r = 0; r < 8; ++r) {
        float v = rm[r];
#pragma unroll
        for (int off = 1; off < 16; off <<= 1)
            v = fmaxf(v, __shfl_xor(v, off, 32));
        if (lr == r) rowmax[b * C + cbase + 8 * half + r] = v;
    }
}

// ---------------- stage 3: q2c_w = softmax_C(rowmax) ------------------------
__global__ void __launch_bounds__(256) k_q2cw(
    const float* __restrict__ rowmax, float* __restrict__ w) {
    const int b = blockIdx.x;
    const int tid = threadIdx.x;
    __shared__ float red[256];
    float vals[4];
    float m = NEG_INF;
#pragma unroll
    for (int j = 0; j < 4; ++j) {
        vals[j] = rowmax[b * C + tid + 256 * j];
        m = fmaxf(m, vals[j]);
    }
    red[tid] = m;
    __syncthreads();
    for (int s = 128; s > 0; s >>= 1) {
        if (tid < s) red[tid] = fmaxf(red[tid], red[tid + s]);
        __syncthreads();
    }
    const float gmax = red[0];
    __syncthreads();
    float e[4];
    float sum = 0.f;
#pragma unroll
    for (int j = 0; j < 4; ++j) { e[j] = __expf(vals[j] - gmax); sum += e[j]; }
    red[tid] = sum;
    __syncthreads();
    for (int s = 128; s > 0; s >>= 1) {
        if (tid < s) red[tid] += red[tid + s];
        __syncthreads();
    }
    const float inv = 1.0f / red[0];
#pragma unroll
    for (int j = 0; j < 4; ++j) w[b * C + tid + 256 * j] = e[j] * inv;
}

// ---------------- stage 4: q2c[b,d] = sum_c w[b,c]*context[b,c,d] -----------
__global__ void __launch_bounds__(256) k_q2c(
    const float* __restrict__ context, const float* __restrict__ w,
    float* __restrict__ q2c) {
    const int b = blockIdx.y;
    const int d = blockIdx.x * 256 + threadIdx.x;
    const float* cp = context + (size_t)b * C * D + d;
    const float* wp = w + b * C;
    float acc = 0.f;
    for (int c = 0; c < C; ++c) {
        __builtin_prefetch(cp + (size_t)(c + 8) * D, 0, 1);
        acc += wp[c] * cp[(size_t)c * D];
    }
    q2c[b * D + d] = acc;
}

// ---------------- stage 5: softmax_q(sim), c2q via WMMA, write all of g -----
// grid: (C/128, B). 8 waves x 16 rows. P tile in LDS (128x128 fp32 = 64KB).
__global__ void __launch_bounds__(256) k_c2q_out(
    const float* __restrict__ sim, const float* __restrict__ query,
    const float* __restrict__ context, const float* __restrict__ q2c,
    float* __restrict__ out) {
    __shared__ float P[128 * 128];
    const int b    = blockIdx.y;
    const int wave = threadIdx.x >> 5;
    const int lane = threadIdx.x & 31;
    const int half = lane >> 4;
    const int lr   = lane & 15;
    const int cbase = blockIdx.x * 128 + wave * 16;
    float* Pw = P + wave * 16 * 128;

    // Phase 1: row softmax over Q into LDS
    for (int r = 0; r < 16; ++r) {
        const float* srow = sim + ((size_t)(b * C) + cbase + r) * Q;
        float v[4];
        float m = NEG_INF;
#pragma unroll
        for (int j = 0; j < 4; ++j) { v[j] = srow[lane + 32 * j]; m = fmaxf(m, v[j]); }
#pragma unroll
        for (int off = 16; off > 0; off >>= 1) m = fmaxf(m, __shfl_xor(m, off, 32));
        float e[4];
        float sum = 0.f;
#pragma unroll
        for (int j = 0; j < 4; ++j) { e[j] = __expf(v[j] - m); sum += e[j]; }
#pragma unroll
        for (int off = 16; off > 0; off >>= 1) sum += __shfl_xor(sum, off, 32);
        const float inv = 1.0f / sum;
#pragma unroll
        for (int j = 0; j < 4; ++j) Pw[r * 128 + lane + 32 * j] = e[j] * inv;
    }
    __syncthreads();

    // Phase 2: c2q tile = P(16x128) x query(128x16) per 16-wide d-tile, then
    // Phase 3: write all 4 output segments for this tile.
    const float* qbase = query + (size_t)(b * Q) * D;
    for (int n0 = 0; n0 < D; n0 += 16) {
        v8f acc = {};
        const float* Prow = Pw + lr * 128 + 2 * half;                 // A frag
        const float* qcol = qbase + (size_t)(2 * half) * D + n0 + lr; // B frag
#pragma unroll 4
        for (int k = 0; k < Q; k += 4) {
            v2f a = *(const v2f*)(Prow + k);
            const float* qp = qcol + (size_t)k * D;
            v2f bf = { qp[0], qp[D] };
            acc = wmma_f32(a, bf, acc);
        }
        const int dcol = n0 + lr;
        const float qcv = q2c[b * D + dcol];
#pragma unroll
        for (int r = 0; r < 8; ++r) {
            const int c = cbase + 8 * half + r;
            const size_t gbase = ((size_t)(b * C) + c) * (size_t)(4 * D);
            const float cval = context[((size_t)(b * C) + c) * D + dcol];
            const float c2qv = acc[r];
            out[gbase + dcol]           = cval;          // context
            out[gbase + D + dcol]       = c2qv;          // c2q
            out[gbase + 2 * D + dcol]   = cval * c2qv;   // context * c2q
            out[gbase + 3 * D + dcol]   = cval * qcv;    // context * q2c
        }
    }
}

extern "C" void kernel_launch(void* const* d_in, const int* in_sizes, int n_in,
                              void* d_out, int out_size, void* d_ws, size_t ws_size,
                              hipStream_t stream) {
    const float* context = (const float*)d_in[0];
    const int*   cmask   = (const int*)d_in[1];
    const float* query   = (const float*)d_in[2];
    const int*   qmask   = (const int*)d_in[3];
    const float* wq      = (const float*)d_in[4];
    const float* wc      = (const float*)d_in[5];
    const float* wqc     = (const float*)d_in[6];
    float* out = (float*)d_out;

    float* ws   = (float*)d_ws;
    float* qw   = ws + OFF_QW;
    float* xq   = ws + OFF_XQ;
    float* xc   = ws + OFF_XC;
    float* rmax = ws + OFF_RM;
    float* w    = ws + OFF_W;
    float* q2c  = ws + OFF_Q2C;
    float* sim  = ws + OFF_SIM;

    k_qw_xq<<<B * Q, 256, 0, stream>>>(query, wq, wqc, qw, xq);
    k_xc<<<B * C, 256, 0, stream>>>(context, wc, xc);
    k_sim<<<dim3(C / 128, B), 256, 0, stream>>>(context, qw, xq, xc,
                                                cmask, qmask, sim, rmax);
    k_q2cw<<<B, 256, 0, stream>>>(rmax, w);
    k_q2c<<<dim3(D / 256, B), 256, 0, stream>>>(context, w, q2c);
    k_c2q_out<<<dim3(C / 128, B), 256, 0, stream>>>(sim, query, context, q2c, out);
}